// EditDistance_46686294507889
// MI455X (gfx1250) — compile-verified
//
#include <hip/hip_runtime.h>

// Problem constants from the reference: ref (512,1024), hyp (512,1024) int32,
// output (1024,) float32.
#define R_DIM 512
#define H_DIM 512
#define B_DIM 1024
#define WAVES_PER_WG 8          // 8 batches per workgroup (256 threads)
#define K_PER_LANE 17           // 17*32 = 544 >= R_DIM+1 = 513 row entries

typedef unsigned int v4u __attribute__((ext_vector_type(4)));
typedef int          v8i __attribute__((ext_vector_type(8)));
typedef int          v4i __attribute__((ext_vector_type(4)));

__device__ __forceinline__ int imin(int a, int b) { return a < b ? a : b; }

// Build a 2D Tensor-DMA descriptor (D# groups 0 and 1) per CDNA5 ISA 8.3/8.4.
//   data_size = 4 bytes, tile = tile_x (contiguous) x tile_y (rows),
//   row stride = stride0 elements, dest = LDS byte offset lds_off.
__device__ __forceinline__ void make_tdm_desc(unsigned long long gaddr,
                                              unsigned lds_off,
                                              unsigned tile_x, unsigned tile_y,
                                              unsigned tensor_x, unsigned tensor_y,
                                              unsigned stride0,
                                              v4u& g0, v8i& g1) {
  // group0: [1:0]=count=1, [63:32]=lds_addr, [120:64]=global_addr, [127:126]=type=2
  g0.x = 1u;
  g0.y = lds_off;
  g0.z = (unsigned)(gaddr & 0xFFFFFFFFull);
  g0.w = ((unsigned)(gaddr >> 32) & 0x01FFFFFFu) | 0x80000000u;
  // group1 word0: workgroup_mask=0, data_size=2 (4B), no atomic/iterate/pad
  g1[0] = (int)(2u << 16);
  // word1: [47:32]=atomic_barrier_addr=0, [63:48]=tensor_dim0[15:0]
  g1[1] = (int)((tensor_x & 0xFFFFu) << 16);
  // word2: tensor_dim0[31:16] | tensor_dim1[15:0]<<16
  g1[2] = (int)((tensor_x >> 16) | ((tensor_y & 0xFFFFu) << 16));
  // word3: tensor_dim1[31:16] | tile_dim0<<16
  g1[3] = (int)((tensor_y >> 16) | ((tile_x & 0xFFFFu) << 16));
  // word4: tile_dim1 | tile_dim2(=0)<<16
  g1[4] = (int)(tile_y & 0xFFFFu);
  // word5: tensor_dim0_stride[31:0]; word6: stride0[47:32] | stride1[15:0]<<16
  g1[5] = (int)stride0;
  g1[6] = 0;
  g1[7] = 0;
}

__global__ __launch_bounds__(WAVES_PER_WG * 32)
void EditDistance_46686294507889_kernel(const int* __restrict__ ref,
                                        const int* __restrict__ hyp,
                                        float* __restrict__ out) {
  // LDS: [0 .. 4095]      hyp tile, token(h, w) at s_lds[h*8 + w]
  //      [4096 .. 8191]   ref tile, token(i, w) at s_lds[4096 + i*8 + w]
  __shared__ int s_lds[8192];

  const int tid  = threadIdx.x;
  const int wave = tid >> 5;
  const int lane = tid & 31;
  const int b0   = blockIdx.x * WAVES_PER_WG;   // first batch of this WG
  const int b    = b0 + wave;                   // this wave's batch element

  // One wave issues two TDM loads: 512x8 int tiles of hyp[:,b0:b0+8] and
  // ref[:,b0:b0+8] (row stride B elements) DMA'd into LDS.
  if (tid < 32) {
    v4u g0; v8i g1;
    v4i gz4 = {0, 0, 0, 0};
    v8i gz8 = {0, 0, 0, 0, 0, 0, 0, 0};
    make_tdm_desc((unsigned long long)(uintptr_t)(hyp + b0), 0u,
                  (unsigned)WAVES_PER_WG, (unsigned)H_DIM,
                  (unsigned)B_DIM, (unsigned)H_DIM, (unsigned)B_DIM, g0, g1);
    __builtin_amdgcn_tensor_load_to_lds(g0, g1, gz4, gz4, gz8, 0);
    make_tdm_desc((unsigned long long)(uintptr_t)(ref + b0), 16384u,
                  (unsigned)WAVES_PER_WG, (unsigned)R_DIM,
                  (unsigned)B_DIM, (unsigned)R_DIM, (unsigned)B_DIM, g0, g1);
    __builtin_amdgcn_tensor_load_to_lds(g0, g1, gz4, gz4, gz8, 0);
    __builtin_amdgcn_s_wait_tensorcnt(0);
  }
  __syncthreads();

  // DP row (length 513, padded to 544) lives in registers: lane L holds
  // entries j = L*17 .. L*17+16. Padded entries use sentinel token -1
  // (never equal to a hyp token in [0,1000)) and never influence j<=512.
  const int base = lane * K_PER_LANE;
  int row[K_PER_LANE];
  int rtok[K_PER_LANE];
#pragma unroll
  for (int k = 0; k < K_PER_LANE; ++k) {
    const int j = base + k;
    row[k]  = j;                                   // row0: i * del_cost
    rtok[k] = (j >= 1 && j <= R_DIM) ? s_lds[4096 + (j - 1) * 8 + wave] : -1;
  }

  const int BIG = 1 << 20;
  for (int h = 0; h < H_DIM; ++h) {
    const int t = s_lds[h * 8 + wave];             // LDS broadcast read
    // previous-row value at j-1 across the lane boundary
    int left = __shfl_up(row[K_PER_LANE - 1], 1, 32);
    if (lane == 0) left = BIG;                     // j==0 has no sub term

    int nw[K_PER_LANE];
    int lm = BIG;                                  // lane-local min of nw[j]-j
#pragma unroll
    for (int k = 0; k < K_PER_LANE; ++k) {
      const int diag = (k == 0) ? left : row[k - 1];
      const int sub  = diag + ((rtok[k] != t) ? 1 : 0);  // substitution/copy
      const int ins  = row[k] + 1;                       // insertion
      const int v    = imin(ins, sub);
      nw[k] = v;
      lm = imin(lm, v - (base + k));
    }

    // Deletion closure: cummin(nw - j) + j == prefix-min across 544 entries.
    // Inclusive wave32 min-scan of per-lane minima -> exclusive prefix.
    int sc = lm;
#pragma unroll
    for (int d = 1; d < 32; d <<= 1) {
      const int u = __shfl_up(sc, d, 32);
      if (lane >= d) sc = imin(sc, u);
    }
    int run = __shfl_up(sc, 1, 32);
    if (lane == 0) run = BIG;
#pragma unroll
    for (int k = 0; k < K_PER_LANE; ++k) {
      run = imin(run, nw[k] - (base + k));
      row[k] = run + base + k;
    }
  }

  // distance = row[R_DIM]; j = 512 -> lane 30 (base 510), k = 2
  if (lane == R_DIM / K_PER_LANE) {
    out[b] = (float)row[R_DIM % K_PER_LANE];
  }
}

extern "C" void kernel_launch(void* const* d_in, const int* in_sizes, int n_in,
                              void* d_out, int out_size, void* d_ws, size_t ws_size,
                              hipStream_t stream) {
  const int* ref = (const int*)d_in[0];   // (R, B) int32
  const int* hyp = (const int*)d_in[1];   // (H, B) int32
  float* out = (float*)d_out;             // (B,) float32
  (void)in_sizes; (void)n_in; (void)out_size; (void)d_ws; (void)ws_size;

  dim3 grid(B_DIM / WAVES_PER_WG);        // 128 workgroups
  dim3 block(WAVES_PER_WG * 32);          // 256 threads = 8 wave32
  EditDistance_46686294507889_kernel<<<grid, block, 0, stream>>>(ref, hyp, out);
}